// GNNEssentials_12584254177944
// MI455X (gfx1250) — compile-verified
//
#include <hip/hip_runtime.h>

#define D 128

typedef float v2f __attribute__((ext_vector_type(2)));
typedef float v8f __attribute__((ext_vector_type(8)));

// Address-space-qualified pointer types for the gfx1250 async LDS-copy builtin.
// Diagnostic from hipcc says param 0 is global (addrspace(1)) `int*`.
typedef __attribute__((address_space(1))) int* as1_iptr;
typedef __attribute__((address_space(3))) int* as3_iptr;

// ---------------------------------------------------------------------------
// Kernel 1: zero the workspace accumulators (agg[N*128] ++ deg[N]).
// ---------------------------------------------------------------------------
__global__ __launch_bounds__(256) void gnn_zero_kernel(float* __restrict__ p,
                                                       long long n) {
  long long i = (long long)blockIdx.x * blockDim.x + threadIdx.x;
  long long stride = (long long)gridDim.x * blockDim.x;
  for (; i < n; i += stride) p[i] = 0.0f;
}

// ---------------------------------------------------------------------------
// Kernel 2: one wave32 per edge.
//   deg[src] += 1           (lane 0)
//   agg[src] += feats[dst]  (32 lanes x float4 = 128 floats, coalesced b128
//                            gather + coalesced f32 atomic stream)
// ---------------------------------------------------------------------------
__global__ __launch_bounds__(256) void gnn_scatter_kernel(
    const float* __restrict__ feats, const long long* __restrict__ edge_index,
    float* __restrict__ agg, float* __restrict__ deg, long long E) {
  long long gtid = (long long)blockIdx.x * blockDim.x + threadIdx.x;
  long long e = gtid >> 5;  // wave per edge (wave32)
  int lane = threadIdx.x & 31;
  if (e >= E) return;

  long long src = edge_index[e];      // row 0 of (2,E)
  long long dst = edge_index[E + e];  // row 1 of (2,E)

  if (lane == 0) atomicAdd(deg + src, 1.0f);

  const float4* f = reinterpret_cast<const float4*>(feats + dst * D);
  float4 v = f[lane];  // 32 lanes * 16B = 512B row, coalesced

  float* a = agg + src * D + (lane << 2);
  atomicAdd(a + 0, v.x);  // -> global_atomic_add_f32
  atomicAdd(a + 1, v.y);
  atomicAdd(a + 2, v.z);
  atomicAdd(a + 3, v.w);
}

// ---------------------------------------------------------------------------
// Kernel 3: out = (agg * (1/deg)) @ W^T + b, masked to 0 where deg == 0.
// fp32 WMMA (V_WMMA_F32_16X16X4_F32). Block = 256 threads = 8 waves.
// Block covers 128 rows; wave w owns rows [r0+16w, r0+16w+16) x all 128 cols.
//
// LDS layout (64 KB): K-pair interleaved transpose of W:
//   Wp[(k>>1)*256 + 2*n + (k&1)] = W[n*128 + k]
// so a lane's B fragment (B[k][n], B[k+1][n]) is ONE aligned ds_load_b64,
// landing directly in a WMMA-ready even VGPR pair.
//
// W staging uses GLOBAL_LOAD_ASYNC_TO_LDS_B32 when available: global side
// coalesced, per-lane LDS dest VGPR absorbs the transpose scatter; synced
// with s_wait_asynccnt 0 + workgroup barrier.
// ---------------------------------------------------------------------------
__global__ __launch_bounds__(256) void gnn_gemm_finalize_kernel(
    const float* __restrict__ agg, const float* __restrict__ deg,
    const float* __restrict__ W, const float* __restrict__ bias,
    float* __restrict__ out, int N) {
  __shared__ float Wp[D * D];  // 64 KB

  int t = threadIdx.x;
#pragma unroll
  for (int i = 0; i < (D * D) / 256; ++i) {
    int flat = i * 256 + t;       // W row-major; consecutive lanes coalesced
    int n = flat >> 7;            // W row (output column)
    int k = flat & (D - 1);       // W col (reduction index)
    int li = (k >> 1) * 256 + (n << 1) + (k & 1);
#if __has_builtin(__builtin_amdgcn_global_load_async_to_lds_b32)
    __builtin_amdgcn_global_load_async_to_lds_b32(
        (as1_iptr)(uintptr_t)(W + flat),
        (as3_iptr)(uintptr_t)(unsigned)(uintptr_t)&Wp[li],
        /*offset=*/0, /*cpol=*/0);
#else
    Wp[li] = W[flat];
#endif
  }
#if __has_builtin(__builtin_amdgcn_global_load_async_to_lds_b32)
#if __has_builtin(__builtin_amdgcn_s_wait_asynccnt)
  __builtin_amdgcn_s_wait_asynccnt(0);
#else
  asm volatile("s_wait_asynccnt 0x0" ::: "memory");
#endif
#endif
  __syncthreads();

  int wave = t >> 5;
  int lane = t & 31;
  int l16 = lane & 15;
  int hi = lane >> 4;  // 0: K pair {0,1}; 1: K pair {2,3} within each 16x4 step

  long long r0 = (long long)blockIdx.x * 128 + (long long)wave * 16;

  // A-fragment source row for this lane (ISA layout: M = lane&15).
  long long rowA = r0 + l16;
  long long rowAc = rowA < N ? rowA : (long long)(N - 1);  // clamp for tail
  float dv = deg[rowAc];
  float invd = dv > 0.0f ? 1.0f / dv : 0.0f;  // fuses the "mean" divide
  const float* aRow = agg + rowAc * D + 2 * hi;

  // Per-lane LDS base: pair row (2*kk + hi), column pair 2*l16.
  const float* wBase = Wp + hi * 256 + 2 * l16;

  v8f acc[8];
#pragma unroll
  for (int ct = 0; ct < 8; ++ct)
#pragma unroll
    for (int j = 0; j < 8; ++j) acc[ct][j] = 0.0f;

  // Software-pipelined K loop: prefetch A for kk+1 before the WMMA chain.
  v2f a_next = *reinterpret_cast<const v2f*>(aRow);  // global_load_b64
#pragma unroll 4
  for (int kk = 0; kk < D / 4; ++kk) {
    v2f a = a_next * invd;  // v_pk_mul_f32
    if (kk < D / 4 - 1)
      a_next = *reinterpret_cast<const v2f*>(aRow + (kk + 1) * 4);
    const float* wrow = wBase + kk * 512;  // two p-rows per 16x4 K-step
#pragma unroll
    for (int ct = 0; ct < 8; ++ct) {
      v2f bf = *reinterpret_cast<const v2f*>(wrow + ct * 32);  // ds_load_b64
      acc[ct] = __builtin_amdgcn_wmma_f32_16x16x4_f32(
          /*neg_a=*/false, a, /*neg_b=*/false, bf,
          /*c_mod=*/(short)0, acc[ct], /*reuse_a=*/false, /*reuse_b=*/false);
    }
  }

  // C/D layout: component v -> row r0 + v + 8*hi, col = ct*16 + (lane&15).
  bool rowok[8];
#pragma unroll
  for (int v = 0; v < 8; ++v) {
    long long row = r0 + v + 8 * hi;
    bool ok = row < N;
    float dg = ok ? deg[row] : 0.0f;
    rowok[v] = ok && (dg > 0.0f);
  }

#pragma unroll
  for (int ct = 0; ct < 8; ++ct) {
    int col = ct * 16 + l16;
    float bcol = bias[col];
#pragma unroll
    for (int v = 0; v < 8; ++v) {
      long long row = r0 + v + 8 * hi;
      if (row < N) {
        out[row * D + col] = rowok[v] ? (acc[ct][v] + bcol) : 0.0f;
      }
    }
  }
}

// ---------------------------------------------------------------------------
// Host launch. Inputs: features [N,128] f32, W [128,128] f32, b [128] f32,
// edge_index [2,E] int64. Output: [N,128] f32.
// Workspace: agg = N*128 floats, deg = N floats (~51.6 MB).
// ---------------------------------------------------------------------------
extern "C" void kernel_launch(void* const* d_in, const int* in_sizes, int n_in,
                              void* d_out, int out_size, void* d_ws,
                              size_t ws_size, hipStream_t stream) {
  const float* feats = (const float*)d_in[0];
  const float* W = (const float*)d_in[1];
  const float* bias = (const float*)d_in[2];
  const long long* edge_index = (const long long*)d_in[3];  // int64 per ref
  float* out = (float*)d_out;

  int N = in_sizes[0] / D;
  long long E = (long long)in_sizes[3] / 2;

  float* agg = (float*)d_ws;
  float* deg = agg + (size_t)N * D;

  long long zn = (long long)N * D + N;
  gnn_zero_kernel<<<1024, 256, 0, stream>>>(agg, zn);

  long long sthreads = E * 32;  // one wave32 per edge
  int sblocks = (int)((sthreads + 255) / 256);
  gnn_scatter_kernel<<<sblocks, 256, 0, stream>>>(feats, edge_index, agg, deg,
                                                  E);

  int gblocks = (N + 127) / 128;
  gnn_gemm_finalize_kernel<<<gblocks, 256, 0, stream>>>(agg, deg, W, bias, out,
                                                        N);
}